// LightGCN_37864431682136
// MI455X (gfx1250) — compile-verified
//
#include <hip/hip_runtime.h>
#include <hip/hip_bf16.h>

// ---------------------------------------------------------------------------
// LightGCN propagation for MI455X (gfx1250, wave32).
//
// final = (x0 + A~x0 + A~^2 x0 + A~^3 x0) / 4,  A~ = D^-1/2 A D^-1/2
//
// The SpMM gather x[col[e],:] is executed by the CDNA5 Tensor Data Mover in
// gather mode: one TENSOR_LOAD_TO_LDS descriptor fetches 8 arbitrary 256B
// embedding rows (32-bit row indices) into LDS, tracked by TENSORcnt and
// double-buffered per wave so DMA overlaps with the atomic scatter. The
// scatter itself uses native global_atomic_add_f32 (L2 atomics); the whole
// ~82 MB working set is resident in the 192 MB L2.
// ---------------------------------------------------------------------------

#define NUM_USERS 100000
#define NUM_ITEMS 50000
#define NNODES    (NUM_USERS + NUM_ITEMS)   // 150000
#define EMB_DIM   64
#define NUM_EDGES 1250000
#define NCHUNK    (NUM_EDGES / 8)           // 156250 8-edge gather chunks

typedef unsigned int v4u __attribute__((ext_vector_type(4)));
typedef int          v4i __attribute__((ext_vector_type(4)));
typedef int          v8i __attribute__((ext_vector_type(8)));

// ---- zero a float buffer (count divisible by 4) ---------------------------
__global__ void lgcn_zero4(float4* __restrict__ p, int n4) {
    int i = blockIdx.x * blockDim.x + threadIdx.x;
    if (i < n4) p[i] = make_float4(0.f, 0.f, 0.f, 0.f);
}

// ---- x0 = concat(user, item); also seed the output accumulator ------------
__global__ void lgcn_init_x(const float4* __restrict__ user,
                            const float4* __restrict__ item,
                            float4* __restrict__ x,
                            float4* __restrict__ out, int n4) {
    int i = blockIdx.x * blockDim.x + threadIdx.x;
    if (i >= n4) return;
    const int userEnd = (NUM_USERS * EMB_DIM) / 4;   // boundary is 16B-aligned
    float4 v = (i < userEnd) ? user[i] : item[i - userEnd];
    x[i] = v;
    out[i] = v;
}

// ---- deg[row[e]] += 1 ------------------------------------------------------
__global__ void lgcn_degree(const long long* __restrict__ ei,
                            float* __restrict__ deg) {
    int e = blockIdx.x * blockDim.x + threadIdx.x;
    if (e >= NUM_EDGES) return;
    int r = (int)ei[e];                  // row = edge_index[0][e]
    atomicAdd(&deg[r], 1.0f);
}

// ---- deg -> deg^-1/2 in place ---------------------------------------------
__global__ void lgcn_inv_sqrt(float* __restrict__ deg, int n) {
    int i = blockIdx.x * blockDim.x + threadIdx.x;
    if (i >= n) return;
    float d = deg[i];
    deg[i] = (d > 0.0f) ? __frsqrt_rn(d) : 0.0f;
}

// ---- norm[e] = dis[row[e]] * dis[col[e]] ----------------------------------
__global__ void lgcn_norm(const long long* __restrict__ ei,
                          const float* __restrict__ dis,
                          float* __restrict__ nrm) {
    int e = blockIdx.x * blockDim.x + threadIdx.x;
    if (e >= NUM_EDGES) return;
    int r = (int)ei[e];
    int c = (int)ei[NUM_EDGES + e];
    nrm[e] = dis[r] * dis[c];
}

// ---------------------------------------------------------------------------
// TDM gather-mode descriptor issue: fetch 8 rows x[col[8c+i], 0:64] (f32)
// into LDS at ldsOff. D# fields per CDNA5 ISA:
//   group0: count=1 | gather_index_size(32b)=1<<30 | gather_mode=1<<31,
//           lds_addr, global_addr[56:0], type=2 in bits[127:126]
//   group1: data_size=2(4B), tensor_dim0=64, tensor_dim1=150000,
//           tile_dim0=64, tile_dim1=8 (#indices), tensor_dim0_stride=64
//   group2/3: row_index_0..7 (32-bit index mode)
// ---------------------------------------------------------------------------
__device__ __forceinline__ void lgcn_issue_gather(const long long* __restrict__ ei,
                                                  const float* __restrict__ x,
                                                  unsigned ldsOff, int chunk) {
    const int b = chunk * 8;
    int c0 = __builtin_amdgcn_readfirstlane((int)ei[NUM_EDGES + b + 0]);
    int c1 = __builtin_amdgcn_readfirstlane((int)ei[NUM_EDGES + b + 1]);
    int c2 = __builtin_amdgcn_readfirstlane((int)ei[NUM_EDGES + b + 2]);
    int c3 = __builtin_amdgcn_readfirstlane((int)ei[NUM_EDGES + b + 3]);
    int c4 = __builtin_amdgcn_readfirstlane((int)ei[NUM_EDGES + b + 4]);
    int c5 = __builtin_amdgcn_readfirstlane((int)ei[NUM_EDGES + b + 5]);
    int c6 = __builtin_amdgcn_readfirstlane((int)ei[NUM_EDGES + b + 6]);
    int c7 = __builtin_amdgcn_readfirstlane((int)ei[NUM_EDGES + b + 7]);

    unsigned long long ga = (unsigned long long)(uintptr_t)x;
    v4u g0 = { 0xC0000001u,                    // count=1, 32b idx, gather mode
               ldsOff,                         // LDS byte address
               (unsigned)ga,                   // global_addr[31:0]
               (unsigned)((ga >> 32) & 0x01FFFFFFu) | 0x80000000u }; // +type=2
    v8i g1 = { 0x00020000,                     // data_size = 4B
               0x00400000,                     // tensor_dim0 = 64
               0x49F00000,                     // tensor_dim1.lo = 150000&0xFFFF
               0x00400002,                     // tensor_dim1.hi=2, tile_dim0=64
               8,                              // tile_dim1 = 8 valid indices
               64,                             // tensor_dim0_stride = 64
               0, 0 };
    v4i g2 = { c0, c1, c2, c3 };
    v4i g3 = { c4, c5, c6, c7 };
#if defined(__clang_major__) && (__clang_major__ >= 23)
    __builtin_amdgcn_tensor_load_to_lds(g0, g1, g2, g3, (v8i){0,0,0,0,0,0,0,0}, 0);
#else
    __builtin_amdgcn_tensor_load_to_lds(g0, g1, g2, g3, 0);
#endif
    __builtin_amdgcn_sched_barrier(0);
}

// ---- consume one gathered 8-row slot: xn[row[e],:] += norm[e]*rows --------
// 4 lanes per edge; each lane owns 16 consecutive floats (4x ds_load_b128,
// 16x global_atomic_add_f32).
__device__ __forceinline__ void lgcn_consume(const char* __restrict__ slot,
                                             const long long* __restrict__ ei,
                                             const float* __restrict__ nrm,
                                             float* __restrict__ xn,
                                             int chunk, int lane) {
    int esub = lane >> 2;                      // 0..7
    int part = lane & 3;                       // 0..3 -> floats [part*16, +16)
    int e    = chunk * 8 + esub;
    int r    = (int)ei[e];
    float w  = nrm[e];
    const char* src = slot + esub * 256 + part * 64;
    float* drow = xn + (size_t)r * EMB_DIM + part * 16;
#pragma unroll
    for (int k = 0; k < 4; ++k) {
        float4 v = *(const float4*)(src + k * 16);   // ds_load_b128
        atomicAdd(drow + k * 4 + 0, v.x * w);
        atomicAdd(drow + k * 4 + 1, v.y * w);
        atomicAdd(drow + k * 4 + 2, v.z * w);
        atomicAdd(drow + k * 4 + 3, v.w * w);
    }
}

// ---- scatter layer: TDM-gather + atomic scatter, double buffered ----------
__global__ void lgcn_scatter_tdm(const long long* __restrict__ ei,
                                 const float* __restrict__ nrm,
                                 const float* __restrict__ x,
                                 float* __restrict__ xn) {
    __shared__ char smem[8 * 2 * 2048];        // 8 waves x 2 slots x 2KB
    const int lane   = threadIdx.x & 31;
    const int waveIb = threadIdx.x >> 5;
    const int wave   = (blockIdx.x * blockDim.x + threadIdx.x) >> 5;
    const int nWaves = (gridDim.x * blockDim.x) >> 5;

    char* slot0 = smem + waveIb * 2 * 2048;
    unsigned off0 = __builtin_amdgcn_readfirstlane((unsigned)(uintptr_t)slot0);

    int c = wave;
    if (c >= NCHUNK) return;

    lgcn_issue_gather(ei, x, off0, c);         // prime slot 0
    int slot = 0;
    int next = c + nWaves;
    for (;;) {
        if (next < NCHUNK) {
            lgcn_issue_gather(ei, x, off0 + (unsigned)((slot ^ 1) * 2048), next);
            __builtin_amdgcn_s_wait_tensorcnt(1);   // oldest gather done
        } else {
            __builtin_amdgcn_s_wait_tensorcnt(0);
        }
        __builtin_amdgcn_sched_barrier(0);
        lgcn_consume(slot0 + slot * 2048, ei, nrm, xn, c, lane);
        if (next >= NCHUNK) break;
        c = next;
        next += nWaves;
        slot ^= 1;
    }
}

// ---- out = (out + xl) * s   (s = 1 for layers 0,1; s = 0.25 fused on last)
__global__ void lgcn_accum(float4* __restrict__ out,
                           const float4* __restrict__ xl,
                           float s, int n4) {
    int i = blockIdx.x * blockDim.x + threadIdx.x;
    if (i >= n4) return;
    float4 a = out[i];
    float4 b = xl[i];
    a.x = (a.x + b.x) * s;
    a.y = (a.y + b.y) * s;
    a.z = (a.z + b.z) * s;
    a.w = (a.w + b.w) * s;
    out[i] = a;
}

// ---------------------------------------------------------------------------
static inline size_t align256(size_t v) { return (v + 255) & ~(size_t)255; }

extern "C" void kernel_launch(void* const* d_in, const int* in_sizes, int n_in,
                              void* d_out, int out_size, void* d_ws, size_t ws_size,
                              hipStream_t stream) {
    (void)in_sizes; (void)n_in; (void)out_size; (void)ws_size;

    const float*     user = (const float*)d_in[0];           // [100000,64] f32
    const float*     item = (const float*)d_in[1];           // [50000,64]  f32
    const long long* ei   = (const long long*)d_in[2];       // [2,1250000] i64
    float*           out  = (float*)d_out;                   // [150000,64] f32

    // ---- carve workspace --------------------------------------------------
    char*  ws   = (char*)d_ws;
    size_t off  = 0;
    float* deg  = (float*)(ws + off); off += align256((size_t)NNODES * 4);          // deg -> dis
    float* nrm  = (float*)(ws + off); off += align256((size_t)NUM_EDGES * 4);       // per-edge norm
    float* xa   = (float*)(ws + off); off += align256((size_t)NNODES * EMB_DIM * 4);
    float* xb   = (float*)(ws + off); off += align256((size_t)NNODES * EMB_DIM * 4);
    // total ~82.4 MB

    const int B = 256;                                   // 8 wave32s per block
    const int nX  = NNODES * EMB_DIM;                    // 9,600,000 floats
    const int nX4 = nX / 4;                              // 2,400,000 float4
    const int nD4 = NNODES / 4;                          // 37,500 float4

    // ---- normalization coefficients (computed once, reused 3 layers) ------
    lgcn_zero4  <<<(nD4 + B - 1) / B, B, 0, stream>>>((float4*)deg, nD4);
    lgcn_degree <<<(NUM_EDGES + B - 1) / B, B, 0, stream>>>(ei, deg);
    lgcn_inv_sqrt<<<(NNODES + B - 1) / B, B, 0, stream>>>(deg, NNODES);
    lgcn_norm   <<<(NUM_EDGES + B - 1) / B, B, 0, stream>>>(ei, deg, nrm);

    // ---- x0 and accumulator ----------------------------------------------
    lgcn_init_x <<<(nX4 + B - 1) / B, B, 0, stream>>>(
        (const float4*)user, (const float4*)item, (float4*)xa, (float4*)out, nX4);

    // ---- 3 propagation layers --------------------------------------------
    float* cur = xa;
    float* nxt = xb;
    const int scatterBlocks = 2048;                      // 16384 waves

    for (int layer = 0; layer < 3; ++layer) {
        lgcn_zero4      <<<(nX4 + B - 1) / B, B, 0, stream>>>((float4*)nxt, nX4);
        lgcn_scatter_tdm<<<scatterBlocks, B, 0, stream>>>(ei, nrm, cur, nxt);
        float s = (layer == 2) ? 0.25f : 1.0f;           // mean over 4 terms fused
        lgcn_accum      <<<(nX4 + B - 1) / B, B, 0, stream>>>(
            (float4*)out, (const float4*)nxt, s, nX4);
        float* tmp = cur; cur = nxt; nxt = tmp;
    }
}